// TankModel_60679297958218
// MI455X (gfx1250) — compile-verified
//
#include <hip/hip_runtime.h>
#include <hip/hip_bf16.h>
#include <stdint.h>

#define TT 64      // timesteps per tile
#define RP 68      // LDS row pitch in floats: 272B, 16B aligned, bank-skewed
#define BT 256     // threads per block == rows per block (8 wave32 -> 2 waves/SIMD)
#define NL (TT/4)  // float4 async loads per thread per tile (16)

typedef float v4f   __attribute__((ext_vector_type(4)));
typedef int   v4i_t __attribute__((vector_size(16)));   // matches builtin's pointee

// ---- CDNA5 async global->LDS copy (ASYNCcnt-tracked) -----------------------
__device__ __forceinline__ void async_copy16(const float* gp, float* lp) {
#if __has_builtin(__builtin_amdgcn_global_load_async_to_lds_b128)
  __builtin_amdgcn_global_load_async_to_lds_b128(
      (__attribute__((address_space(1))) v4i_t*)const_cast<float*>(gp),
      (__attribute__((address_space(3))) v4i_t*)lp, 0, 0);
#else
  unsigned laddr =
      (unsigned)(uintptr_t)(__attribute__((address_space(3))) float*)lp;
  asm volatile("global_load_async_to_lds_b128 %0, %1, off"
               :: "v"(laddr), "v"((const void*)gp) : "memory");
#endif
}

template <int N>
__device__ __forceinline__ void wait_async() {
#if __has_builtin(__builtin_amdgcn_s_wait_asynccnt)
  __builtin_amdgcn_s_wait_asynccnt(N);
  asm volatile("" ::: "memory");
#else
  asm volatile("s_wait_asynccnt %0" :: "i"(N) : "memory");
#endif
}

__global__ __launch_bounds__(BT) void tank_scan_kernel(
    const float* __restrict__ P, const int* __restrict__ len_p,
    const float* __restrict__ S0, const float* __restrict__ area_p,
    const float* __restrict__ r1h1_p, const float* __restrict__ r1h2_p,
    const float* __restrict__ h1h1_p, const float* __restrict__ h1h2_p,
    const float* __restrict__ ir2_p,  const float* __restrict__ or2_p,
    const float* __restrict__ h2_p,
    const float* __restrict__ ir3_p,  const float* __restrict__ or3_p,
    const float* __restrict__ h3_p,
    const float* __restrict__ ir4_p,  const float* __restrict__ or4_p,
    float* __restrict__ Out, int B, int T)
{
  __shared__ float bufA[BT * RP];
  __shared__ float bufB[BT * RP];

  const int tid  = threadIdx.x;
  const int b0   = blockIdx.x * BT;
  const int b    = b0 + tid;
  const bool rowok = (b < B);
  const int rows = (B - b0 < BT) ? (B - b0) : BT;
  const int len  = rowok ? len_p[b] : 0;

  // uniform scalars (SMEM loads)
  const float r1h1 = r1h1_p[0], r1h2 = r1h2_p[0];
  const float h1h1 = h1h1_p[0], h1h2 = h1h2_p[0];
  const float ir2  = ir2_p[0],  or2  = or2_p[0],  h2 = h2_p[0];
  const float ir3  = ir3_p[0],  or3  = or3_p[0],  h3 = h3_p[0];
  const float ir4  = ir4_p[0],  or4  = or4_p[0];
  const float scale = area_p[0] / 3.6f;

  float w1 = S0[0], w2 = S0[1], w3 = S0[2], w4 = S0[3];

  const int ntiles = (T + TT - 1) / TT;

  // one tank-model step, reference op order preserved
  auto step = [&](float p, float m) -> float {
    float of1h1 = fmaxf(r1h1 * (w1 - h1h1 + p), 0.0f) * m;
    float of1h2 = fmaxf(r1h2 * (w1 - h1h2 + p), 0.0f) * m;
    float of1   = (of1h1 + of1h2) * m;
    float out12 = fmaxf(ir2 * (w1 + p), 0.0f) * m;
    w1 = w1 + (p - of1 - out12) * m;
    float of2   = fmaxf(or2 * (w2 + out12 - h2), 0.0f) * m;
    float out23 = fmaxf(ir3 * (w2 + out12), 0.0f) * m;
    w2 = w2 + (out12 - of2 - out23) * m;
    float of3   = fmaxf(or3 * (w3 + out23 - h3), 0.0f) * m;
    float out34 = fmaxf(ir4 * (w3 + out23), 0.0f) * m;
    w3 = w3 + (out23 - of3 - out34) * m;
    float of4   = fmaxf(or4 * (w4 + out34), 0.0f) * m;
    w4 = w4 + (out34 - of4) * m;
    return (of1 + of2 + of3 + of4) * m * scale;
  };

  auto compute_tile = [&](float* cur, int t0, int tmax) {
    if (rowok) {
      float* prow = cur + tid * RP;           // this thread's row (in-place)
#pragma unroll 4
      for (int t = 0; t < tmax; t += 4) {
        v4f pv = *(const v4f*)(prow + t);
        v4f tv;
        tv.x = step(pv.x, ((t0 + t + 0) < len) ? 1.0f : 0.0f);
        tv.y = step(pv.y, ((t0 + t + 1) < len) ? 1.0f : 0.0f);
        tv.z = step(pv.z, ((t0 + t + 2) < len) ? 1.0f : 0.0f);
        tv.w = step(pv.w, ((t0 + t + 3) < len) ? 1.0f : 0.0f);
        *(v4f*)(prow + t) = tv;               // overwrite input tile in place
      }
    }
  };

  // ---- fast path: every lane issues exactly NL loads per tile -------------
  auto issue_full = [&](int t0, float* lbuf) {
    const float* gbase = P + (size_t)b0 * T + t0;
#pragma unroll
    for (int k = 0; k < NL; ++k) {
      int idx = k * BT + tid;       // 256x16 float4 tile
      int row = idx >> 4;
      int c   = (idx & 15) << 2;
      async_copy16(gbase + (size_t)row * T + c, lbuf + row * RP + c);
    }
  };

  auto issue_guard = [&](int t0, float* lbuf) {
    const int tcols = (T - t0 < TT) ? (T - t0) : TT;
    const float* gbase = P + (size_t)b0 * T + t0;
#pragma unroll
    for (int k = 0; k < NL; ++k) {
      int idx = k * BT + tid;
      int row = idx >> 4;
      int c   = (idx & 15) << 2;
      if (row < rows && (c + 4) <= tcols)
        async_copy16(gbase + (size_t)row * T + c, lbuf + row * RP + c);
    }
  };

  const bool full = (rows == BT) && ((T % TT) == 0);

  if (full) {
    issue_full(0, bufA);
    if (ntiles > 1) issue_full(TT, bufB);
    for (int i = 0; i < ntiles; ++i) {
      float* cur = (i & 1) ? bufB : bufA;
      const int t0 = i * TT;
      if (i + 1 < ntiles) wait_async<NL>();   // retire oldest tile only
      else                wait_async<0>();
      __syncthreads();                        // tile i visible block-wide
      compute_tile(cur, t0, TT);
      __syncthreads();                        // totals staged
#pragma unroll
      for (int k = 0; k < NL; ++k) {          // guard-free coalesced NT store
        int idx = k * BT + tid;
        int row = idx >> 4;
        int c   = (idx & 15) << 2;
        v4f v = *(const v4f*)(cur + row * RP + c);
        __builtin_nontemporal_store(
            v, (v4f*)(Out + (size_t)(b0 + row) * T + t0 + c));
      }
      __syncthreads();                        // cur free for reload
      if (i + 2 < ntiles) issue_full((i + 2) * TT, cur);
    }
  } else {
    // conservative path for ragged shapes: drain fully before reuse
    issue_guard(0, bufA);
    for (int i = 0; i < ntiles; ++i) {
      float* cur = (i & 1) ? bufB : bufA;
      float* nxt = (i & 1) ? bufA : bufB;
      const int t0   = i * TT;
      const int tmax = (T - t0 < TT) ? (T - t0) : TT;
      wait_async<0>();
      __syncthreads();
      if (i + 1 < ntiles) issue_guard(t0 + TT, nxt);
      compute_tile(cur, t0, tmax);
      __syncthreads();
#pragma unroll
      for (int k = 0; k < NL; ++k) {
        int idx = k * BT + tid;
        int row = idx >> 4;
        int c   = (idx & 15) << 2;
        if (row < rows && (c + 4) <= tmax) {
          v4f v = *(const v4f*)(cur + row * RP + c);
          __builtin_nontemporal_store(
              v, (v4f*)(Out + (size_t)(b0 + row) * T + t0 + c));
        }
      }
      __syncthreads();
    }
  }
}

extern "C" void kernel_launch(void* const* d_in, const int* in_sizes, int n_in,
                              void* d_out, int out_size, void* d_ws, size_t ws_size,
                              hipStream_t stream) {
  const float* P       = (const float*)d_in[0];
  const int*   lengths = (const int*)  d_in[1];
  const float* S0      = (const float*)d_in[2];
  const float* Area    = (const float*)d_in[3];
  // d_in[4] = baseflow (unused by forward)
  const int B = in_sizes[1];
  const int T = in_sizes[0] / B;

  dim3 grid((B + BT - 1) / BT), block(BT);
  tank_scan_kernel<<<grid, block, 0, stream>>>(
      P, lengths, S0, Area,
      (const float*)d_in[5],  (const float*)d_in[6],
      (const float*)d_in[7],  (const float*)d_in[8],
      (const float*)d_in[9],  (const float*)d_in[10], (const float*)d_in[11],
      (const float*)d_in[12], (const float*)d_in[13], (const float*)d_in[14],
      (const float*)d_in[15], (const float*)d_in[16],
      (float*)d_out, B, T);
}